// CrossGAT_71871982731539
// MI455X (gfx1250) — compile-verified
//
#include <hip/hip_runtime.h>
#include <hip/hip_bf16.h>
#include <math.h>

#define NDIM 128
#define NEG_SLOPE 0.2f

typedef __attribute__((ext_vector_type(2))) float v2f;
typedef __attribute__((ext_vector_type(8))) float v8f;

// ---------- atomics ----------
__device__ __forceinline__ void atomic_max_f32(float* addr, float v) {
  // exact float max via sign-split integer atomics (init must be -inf)
  if (v >= 0.0f) {
    atomicMax((int*)addr, __float_as_int(v));
  } else {
    atomicMin((unsigned int*)addr, __float_as_uint(v));
  }
}

__device__ __forceinline__ void atomic_add_f32(float* addr, float v) {
  unsafeAtomicAdd(addr, v);  // lowers to global_atomic_add_f32 on gfx1250
}

// ---------- 1) H[N,128] = X[N,128] @ W[128,128] via V_WMMA_F32_16X16X4_F32 ----------
// Block = 256 threads = 8 waves. Block handles 16 rows; wave w handles cols [16w,16w+16).
__global__ __launch_bounds__(256) void k_gemm(const float* __restrict__ X,
                                              const float* __restrict__ W,
                                              float* __restrict__ H, int N) {
  const int wave = threadIdx.x >> 5;
  const int lane = threadIdx.x & 31;
  const int half = lane >> 4;   // 0: lanes 0-15, 1: lanes 16-31
  const int l15  = lane & 15;
  const int row0 = blockIdx.x * 16;
  const int col0 = wave * 16;
  if (row0 >= N) return;

  const float* xrow = X + (size_t)(row0 + l15) * NDIM;  // A: lane 0-15 <-> M, dup in upper half
  const float* wcol = W + col0 + l15;                   // B: lane 0-15 <-> N, dup in upper half
  v8f acc = {};
  for (int k = 0; k < NDIM; k += 4) {
    const int ka = k + half * 2;  // A/B VGPR0: K=ka, VGPR1: K=ka+1 (ISA 16x4 f32 layout)
    v2f a = *(const v2f*)(xrow + ka);
    v2f b;
    b.x = wcol[(size_t)ka * NDIM];
    b.y = wcol[(size_t)(ka + 1) * NDIM];
    acc = __builtin_amdgcn_wmma_f32_16x16x4_f32(false, a, false, b,
                                                (short)0, acc, false, false);
  }
  // D layout: VGPR r -> M = r + 8*half, N = l15
  float* hp = H + (size_t)row0 * NDIM + col0 + l15;
#pragma unroll
  for (int r = 0; r < 8; ++r)
    hp[(size_t)(r + half * 8) * NDIM] = acc[r];
}

// ---------- 2) al/ar dots + init m,s + zero accumulator row ----------
__global__ __launch_bounds__(256) void k_dots_init(const float* __restrict__ H,
    const float* __restrict__ a_s, const float* __restrict__ a_d,
    float* __restrict__ al, float* __restrict__ ar,
    float* __restrict__ m, float* __restrict__ s,
    float* __restrict__ acc, int N) {
  const int w    = (int)((blockIdx.x * (size_t)blockDim.x + threadIdx.x) >> 5);
  const int lane = threadIdx.x & 31;
  if (w >= N) return;
  const float4 hv = ((const float4*)(H + (size_t)w * NDIM))[lane];
  const float4 sv = ((const float4*)a_s)[lane];
  const float4 dv = ((const float4*)a_d)[lane];
  float da = hv.x * sv.x + hv.y * sv.y + hv.z * sv.z + hv.w * sv.w;
  float db = hv.x * dv.x + hv.y * dv.y + hv.z * dv.z + hv.w * dv.w;
#pragma unroll
  for (int off = 16; off > 0; off >>= 1) {
    da += __shfl_xor(da, off, 32);
    db += __shfl_xor(db, off, 32);
  }
  ((float4*)(acc + (size_t)w * NDIM))[lane] = make_float4(0.f, 0.f, 0.f, 0.f);
  if (lane == 0) {
    al[w] = da;
    ar[w] = db;
    m[w]  = -__builtin_huge_valf();
    s[w]  = 0.0f;
  }
}

// ---------- 3) segment max over destinations ----------
__global__ __launch_bounds__(256) void k_edge_max(const int* __restrict__ src,
    const int* __restrict__ dst, const float* __restrict__ al,
    const float* __restrict__ ar, float* __restrict__ m, int E, int N) {
  const int i = blockIdx.x * blockDim.x + threadIdx.x;
  if (i >= E + N) return;
  int s_, d_;
  if (i < E) { s_ = src[i]; d_ = dst[i]; } else { s_ = i - E; d_ = s_; }  // self-loops
  float v = al[s_] + ar[d_];
  v = (v > 0.0f) ? v : v * NEG_SLOPE;
  atomic_max_f32(m + d_, v);
}

// ---------- 4) exp(e - m[dst]) per edge + segment sum ----------
__global__ __launch_bounds__(256) void k_edge_expsum(const int* __restrict__ src,
    const int* __restrict__ dst, const float* __restrict__ al,
    const float* __restrict__ ar, const float* __restrict__ m,
    float* __restrict__ ex, float* __restrict__ ssum, int E, int N) {
  const int i = blockIdx.x * blockDim.x + threadIdx.x;
  if (i >= E + N) return;
  int s_, d_;
  if (i < E) { s_ = src[i]; d_ = dst[i]; } else { s_ = i - E; d_ = s_; }
  float v = al[s_] + ar[d_];
  v = (v > 0.0f) ? v : v * NEG_SLOPE;
  const float e = __expf(v - m[d_]);
  ex[i] = e;
  atomic_add_f32(ssum + d_, e);
}

// ---------- 5) acc[dst] += alpha * h[src]  (wave per edge, float4 per lane) ----------
__global__ __launch_bounds__(256) void k_aggregate(const int* __restrict__ src,
    const int* __restrict__ dst, const float* __restrict__ H,
    const float* __restrict__ ex, const float* __restrict__ ssum,
    float* __restrict__ acc, int E, int N) {
  const int w    = (int)((blockIdx.x * (size_t)blockDim.x + threadIdx.x) >> 5);
  const int lane = threadIdx.x & 31;
  if (w >= E + N) return;
  int s_, d_;
  if (w < E) { s_ = src[w]; d_ = dst[w]; } else { s_ = w - E; d_ = s_; }
  const float alpha = ex[w] / ssum[d_];
  const float4 hv = ((const float4*)(H + (size_t)s_ * NDIM))[lane];
  float* ap = acc + (size_t)d_ * NDIM + lane * 4;
  atomic_add_f32(ap + 0, hv.x * alpha);
  atomic_add_f32(ap + 1, hv.y * alpha);
  atomic_add_f32(ap + 2, hv.z * alpha);
  atomic_add_f32(ap + 3, hv.w * alpha);
}

// ---------- 6) out = acc + b, optional ReLU ----------
__global__ __launch_bounds__(256) void k_bias_act(const float* __restrict__ acc,
    const float* __restrict__ b, float* __restrict__ out, int total, int relu) {
  const int i = blockIdx.x * blockDim.x + threadIdx.x;
  if (i >= total) return;
  float v = acc[i] + b[i & (NDIM - 1)];
  if (relu) v = fmaxf(v, 0.0f);
  out[i] = v;
}

extern "C" void kernel_launch(void* const* d_in, const int* in_sizes, int n_in,
                              void* d_out, int out_size, void* d_ws, size_t ws_size,
                              hipStream_t stream) {
  const float* x   = (const float*)d_in[0];
  const int*   ei  = (const int*)d_in[21];
  const int*   eic = (const int*)d_in[22];
  const int E = in_sizes[21] / 2;
  const int N = in_sizes[0] / NDIM;  // 50000
  const size_t nf = (size_t)N * NDIM;

  // workspace layout (floats): H | ACC | XB | al | ar | m | s | ex  (~81 MB)
  float* ws  = (float*)d_ws;
  float* H   = ws;
  float* ACC = ws + nf;
  float* XB  = ws + 2 * nf;
  float* AL  = ws + 3 * nf;
  float* AR  = AL + N;
  float* M   = AR + N;
  float* S   = M + N;
  float* EX  = S + N;

  const float* cur = x;
  for (int l = 0; l < 5; ++l) {
    const float* W   = (const float*)d_in[1 + 4 * l];
    const float* a_s = (const float*)d_in[2 + 4 * l];
    const float* a_d = (const float*)d_in[3 + 4 * l];
    const float* b   = (const float*)d_in[4 + 4 * l];
    const int* e     = (l & 1) ? eic : ei;  // layers 1,3,5 -> ei; 2,4 -> cross
    const int* esrc  = e;
    const int* edst  = e + E;
    const int tot    = E + N;

    k_gemm<<<(N + 15) / 16, 256, 0, stream>>>(cur, W, H, N);
    k_dots_init<<<(int)((nf * 32 / NDIM + 255) / 256), 256, 0, stream>>>(
        H, a_s, a_d, AL, AR, M, S, ACC, N);
    k_edge_max<<<(tot + 255) / 256, 256, 0, stream>>>(esrc, edst, AL, AR, M, E, N);
    k_edge_expsum<<<(tot + 255) / 256, 256, 0, stream>>>(esrc, edst, AL, AR, M, EX, S, E, N);
    k_aggregate<<<(int)(((size_t)tot * 32 + 255) / 256), 256, 0, stream>>>(
        esrc, edst, H, EX, S, ACC, E, N);

    float* dstbuf = (l == 4) ? (float*)d_out : XB;
    k_bias_act<<<(int)((nf + 255) / 256), 256, 0, stream>>>(ACC, b, dstbuf, (int)nf,
                                                            (l < 4) ? 1 : 0);
    cur = XB;
  }
}